// SBGNN_19542101197290
// MI455X (gfx1250) — compile-verified
//
#include <hip/hip_runtime.h>
#include <hip/hip_bf16.h>
#include <math.h>

// SBGNN layer for MI455X (gfx1250, wave32).
// N=50000 per side, EMB=35, DEG=32 (== wave32: one lane per edge).
#define NN   50000
#define EMB  35
#define DEG  32
#define H1   70     // 2*EMB
#define KIN  105    // 3*EMB
#define XS   108    // padded concat width (3*EMB -> 108, /4 and 8B-aligned rows)
#define VS   36     // padded new_emb row stride (8B-aligned rows)
#define W0R  36     // Wp rows  (35 -> 36)
#define W0C  48     // Wp cols  (35 -> 48, 3 N-tiles)
#define W1R  108    // W1p rows (105 -> 108)
#define W1C  80     // W1p cols (70 -> 80, 5 N-tiles)
#define W2R  72     // W2p rows (70 -> 72)
#define W2C  48     // W2p cols (35 -> 48, 3 N-tiles)

typedef float v2f __attribute__((ext_vector_type(2)));
typedef float v8f __attribute__((ext_vector_type(8)));

// D = A(16x4 f32) * B(4x16 f32) + C(16x16 f32), full-precision CDNA5 WMMA.
static __device__ __forceinline__ v8f wmma_f32_k4(v2f a, v2f b, v8f c) {
  return __builtin_amdgcn_wmma_f32_16x16x4_f32(
      false, a, false, b, (short)0, c, false, false);
}

static __device__ __forceinline__ float waveReduceSum(float v) {
#pragma unroll
  for (int off = 16; off > 0; off >>= 1) v += __shfl_xor(v, off, 32);
  return v;
}

// ---------------------------------------------------------------------------
// Generic zero-padding copy: dst[drows][dcols] <- src[srows][scols], pad = 0.
// Used once per call to build padded weights / activations in workspace.
// ---------------------------------------------------------------------------
__global__ __launch_bounds__(256) void padcopy(const float* __restrict__ src,
                                               int srows, int scols,
                                               float* __restrict__ dst,
                                               int drows, int dcols) {
  const int idx = blockIdx.x * blockDim.x + threadIdx.x;
  if (idx >= drows * dcols) return;
  const int r = idx / dcols, c = idx - r * dcols;
  dst[idx] = (r < srows && c < scols) ? src[r * scols + c] : 0.f;
}

// ---------------------------------------------------------------------------
// Kernel 1: new_emb[n][VS] = xp[:, 0:35] @ W + bias.
// A from padded xp (stride 108, b64 loads), B/bias from padded globals.
// No LDS, no guards in the K loop; 27 WMMAs per wave.
// ---------------------------------------------------------------------------
__global__ __launch_bounds__(256) void newemb_gemm(const float* __restrict__ xp,
                                                   const float* __restrict__ Wp,
                                                   const float* __restrict__ bp,
                                                   float* __restrict__ outTab,
                                                   int n) {
  const int wave = threadIdx.x >> 5, lane = threadIdx.x & 31;
  const int half = lane >> 4, l16 = lane & 15;
  const int node0 = (blockIdx.x * 8 + wave) * 16;
  if (node0 >= n) return;                       // wave-uniform exit
  int nodeA = node0 + l16;
  nodeA = nodeA < n ? nodeA : n - 1;            // branch-free clamp
  const float* arow = xp + (size_t)nodeA * XS;  // cols 0..35 (35 is pad/any)

  v8f z = {};
  v8f acc[3];
  acc[0] = z; acc[1] = z; acc[2] = z;

#pragma unroll
  for (int kk = 0; kk < 36; kk += 4) {
    const int k0 = kk + 2 * half;               // even -> 8B aligned
    const v2f a = *(const v2f*)(arow + k0);
#pragma unroll
    for (int t = 0; t < 3; t++) {
      const int nn = t * 16 + l16;
      v2f b;
      b.x = Wp[k0 * W0C + nn];
      b.y = Wp[(k0 + 1) * W0C + nn];
      acc[t] = wmma_f32_k4(a, b, acc[t]);
    }
  }

#pragma unroll
  for (int t = 0; t < 3; t++) {
#pragma unroll
    for (int r = 0; r < 8; r++) {
      const int node = node0 + r + 8 * half;
      const int nn   = t * 16 + l16;
      if (node < n && nn < EMB) outTab[(size_t)node * VS + nn] = acc[t][r] + bp[nn];
    }
  }
}

// ---------------------------------------------------------------------------
// Kernel 2: out[i] = dot(tab[i, 0:35], att[0:35])   (wave per node)
// ---------------------------------------------------------------------------
__global__ __launch_bounds__(256) void rowdot(const float* __restrict__ tab,
                                              int stride,
                                              const float* __restrict__ att,
                                              float* __restrict__ out, int n) {
  const int node = blockIdx.x * (blockDim.x >> 5) + (threadIdx.x >> 5);
  if (node >= n) return;
  const int lane = threadIdx.x & 31;
  const float* row = tab + (size_t)node * stride;
  float p = row[lane] * att[lane];
  if (lane < EMB - 32) p += row[32 + lane] * att[32 + lane];
  const float s = waveReduceSum(p);
  if (lane == 0) out[node] = s;
}

// ---------------------------------------------------------------------------
// Kernel 3: attention aggregation. Wave per node; lane l owns edge node*32+l.
// Writes the message straight into xp[node][obase .. obase+34].
// Fixed xor-reduction order => deterministic.
// ---------------------------------------------------------------------------
__global__ __launch_bounds__(256) void edge_agg(const int* __restrict__ dst,
                                                const float* __restrict__ s_src,
                                                const float* __restrict__ s_dst,
                                                const float* __restrict__ vtab,
                                                float* __restrict__ outx,
                                                int obase, int n) {
  const int node = blockIdx.x * (blockDim.x >> 5) + (threadIdx.x >> 5);
  if (node >= n) return;
  const int lane = threadIdx.x & 31;

  const int d    = dst[node * DEG + lane];
  const float t  = s_src[node] + s_dst[d];
  const float el = (t > 0.f) ? t : 0.1f * expm1f(t);
  const float w  = expf(el);

  const float den = waveReduceSum(w);
  const float inv = 1.f / ((den == 0.f) ? 1.f : den);

  const float* row = vtab + (size_t)d * VS;     // 8B-aligned row
  float r[EMB];
#pragma unroll
  for (int k = 0; k < 17; k++) {                // 17x b64 + 1x b32 gathers
    const v2f q = *(const v2f*)(row + 2 * k);
    r[2 * k] = q.x; r[2 * k + 1] = q.y;
  }
  r[34] = row[34];

  float o_lo = 0.f, o_hi = 0.f;
#pragma unroll
  for (int k = 0; k < EMB; k++) {
    const float s = waveReduceSum(w * r[k]);
    if (k < 32) { if (lane == k)      o_lo = s; }
    else        { if (lane == k - 32) o_hi = s; }
  }
  float* orow = outx + (size_t)node * XS + obase;
  orow[lane] = o_lo * inv;                          // dims 0..31
  if (lane < EMB - 32) orow[32 + lane] = o_hi * inv; // dims 32..34
}

// ---------------------------------------------------------------------------
// Kernel 4: fused 2-layer MLP on padded x (K=105->108) -> PReLU -> (K=70->72).
// All fragment loads unguarded (padded operands); only final store is guarded.
// 135 + 54 WMMAs per wave.
// ---------------------------------------------------------------------------
__global__ __launch_bounds__(256) void update_mlp(const float* __restrict__ xp,
                                                  const float* __restrict__ W1p,
                                                  const float* __restrict__ b1p,
                                                  const float* __restrict__ alphap,
                                                  const float* __restrict__ W2p,
                                                  const float* __restrict__ b2p,
                                                  float* __restrict__ out, int n) {
  __shared__ __attribute__((aligned(16))) float hbuf[8][16 * W1C];

  const int wave = threadIdx.x >> 5, lane = threadIdx.x & 31;
  const int half = lane >> 4, l16 = lane & 15;
  const int node0 = (blockIdx.x * 8 + wave) * 16;
  int nodeA = node0 + l16;
  nodeA = nodeA < n ? nodeA : n - 1;            // clamp; no early return (barrier)
  const float alpha = alphap[0];
  const float* arow = xp + (size_t)nodeA * XS;

  v8f z = {};
  v8f acc[5];
#pragma unroll
  for (int t = 0; t < 5; t++) acc[t] = z;

  // ---- layer 1: [16x108] @ [108x80] ----
#pragma unroll 3
  for (int kk = 0; kk < W1R; kk += 4) {
    const int k0 = kk + 2 * half;
    const v2f a = *(const v2f*)(arow + k0);
#pragma unroll
    for (int t = 0; t < 5; t++) {
      const int nn = t * 16 + l16;
      v2f b;
      b.x = W1p[k0 * W1C + nn];
      b.y = W1p[(k0 + 1) * W1C + nn];
      acc[t] = wmma_f32_k4(a, b, acc[t]);
    }
  }

  // bias + PReLU; pad cols (>=70) come out exactly 0 (0 + 0 bias, prelu(0)=0)
  float* hrow = &hbuf[wave][0];
#pragma unroll
  for (int t = 0; t < 5; t++) {
#pragma unroll
    for (int r = 0; r < 8; r++) {
      const int m  = r + 8 * half;
      const int nn = t * 16 + l16;
      float h = acc[t][r] + b1p[nn];
      h = (h > 0.f) ? h : alpha * h;
      hrow[m * W1C + nn] = h;
    }
  }
  __syncthreads();

  // ---- layer 2: [16x72] @ [72x48] ----
  v8f acc2[3];
  acc2[0] = z; acc2[1] = z; acc2[2] = z;
#pragma unroll 3
  for (int kk = 0; kk < W2R; kk += 4) {
    const int k0 = kk + 2 * half;
    const v2f a = *(const v2f*)(hrow + l16 * W1C + k0);
#pragma unroll
    for (int t = 0; t < 3; t++) {
      const int nn = t * 16 + l16;
      v2f b;
      b.x = W2p[k0 * W2C + nn];
      b.y = W2p[(k0 + 1) * W2C + nn];
      acc2[t] = wmma_f32_k4(a, b, acc2[t]);
    }
  }

#pragma unroll
  for (int t = 0; t < 3; t++) {
#pragma unroll
    for (int r = 0; r < 8; r++) {
      const int node = node0 + r + 8 * half;
      const int nn   = t * 16 + l16;
      if (node < n && nn < EMB) out[(size_t)node * EMB + nn] = acc2[t][r] + b2p[nn];
    }
  }
}

// ---------------------------------------------------------------------------
// Host launcher
// ---------------------------------------------------------------------------
extern "C" void kernel_launch(void* const* d_in, const int* in_sizes, int n_in,
                              void* d_out, int out_size, void* d_ws, size_t ws_size,
                              hipStream_t stream) {
  const float* fa = (const float*)d_in[0];
  const float* fb = (const float*)d_in[1];
  const int* dstv[4] = {(const int*)d_in[2], (const int*)d_in[3],
                        (const int*)d_in[4], (const int*)d_in[5]};
  const float* Wt[4]   = {(const float*)d_in[6],  (const float*)d_in[9],
                          (const float*)d_in[12], (const float*)d_in[15]};
  const float* bt[4]   = {(const float*)d_in[7],  (const float*)d_in[10],
                          (const float*)d_in[13], (const float*)d_in[16]};
  const float* attv[4] = {(const float*)d_in[8],  (const float*)d_in[11],
                          (const float*)d_in[14], (const float*)d_in[17]};
  const float* W1    = (const float*)d_in[18];
  const float* b1    = (const float*)d_in[19];
  const float* alpha = (const float*)d_in[20];
  const float* W2    = (const float*)d_in[21];
  const float* b2    = (const float*)d_in[22];

  const int N = NN;
  float* ws = (float*)d_ws;
  size_t off = 0;
  float* newemb[4];
  for (int a = 0; a < 4; a++) { newemb[a] = ws + off; off += (size_t)N * VS; }
  float* ssrc = ws + off; off += (size_t)4 * N;
  float* sdst = ws + off; off += (size_t)4 * N;
  float* xpa  = ws + off; off += (size_t)N * XS;
  float* xpb  = ws + off; off += (size_t)N * XS;
  float* Wp[4]; float* bp[4];
  for (int a = 0; a < 4; a++) { Wp[a] = ws + off; off += W0R * W0C; }
  for (int a = 0; a < 4; a++) { bp[a] = ws + off; off += W0C; }
  float* W1p = ws + off; off += W1R * W1C;
  float* b1p = ws + off; off += W1C;
  float* W2p = ws + off; off += W2R * W2C;
  float* b2p = ws + off; off += W2C;

  dim3 blk(256);
  auto pgrid = [](int total) { return dim3((total + 255) / 256); };

  // 1) pad weights + biases into workspace
  for (int a = 0; a < 4; a++) {
    padcopy<<<pgrid(W0R * W0C), blk, 0, stream>>>(Wt[a], EMB, EMB, Wp[a], W0R, W0C);
    padcopy<<<pgrid(W0C), blk, 0, stream>>>(bt[a], 1, EMB, bp[a], 1, W0C);
  }
  padcopy<<<pgrid(W1R * W1C), blk, 0, stream>>>(W1, KIN, H1, W1p, W1R, W1C);
  padcopy<<<pgrid(W1C), blk, 0, stream>>>(b1, 1, H1, b1p, 1, W1C);
  padcopy<<<pgrid(W2R * W2C), blk, 0, stream>>>(W2, H1, EMB, W2p, W2R, W2C);
  padcopy<<<pgrid(W2C), blk, 0, stream>>>(b2, 1, EMB, b2p, 1, W2C);

  // 2) build padded concat buffers: cols 0..34 = features, rest zeroed
  //    (cols 35..104 are later overwritten by edge_agg messages)
  padcopy<<<pgrid(N * XS), blk, 0, stream>>>(fa, N, EMB, xpa, N, XS);
  padcopy<<<pgrid(N * XS), blk, 0, stream>>>(fb, N, EMB, xpb, N, XS);

  const int gridG = (N + 127) / 128;  // 8 waves x 16 nodes
  const int gridW = (N + 7) / 8;      // 8 waves x 1 node

  // 3) new_emb = fb@W+b (aggs 0,1) / fa@W+b (aggs 2,3), via WMMA
  const float* xsrcDst[4] = {xpb, xpb, xpa, xpa};  // feature side being transformed
  for (int a = 0; a < 4; a++)
    newemb_gemm<<<gridG, blk, 0, stream>>>(xsrcDst[a], Wp[a], bp[a], newemb[a], N);

  // 4) attention logits s_a (from raw features) and s_b (from new_emb)
  const float* featSrc[4] = {fa, fa, fb, fb};
  for (int a = 0; a < 4; a++) {
    rowdot<<<gridW, blk, 0, stream>>>(featSrc[a], EMB, attv[a], ssrc + (size_t)a * N, N);
    rowdot<<<gridW, blk, 0, stream>>>(newemb[a], VS, attv[a] + EMB, sdst + (size_t)a * N, N);
  }

  // 5) gather + softmax-style aggregation, writing messages into xp cols
  float* xdst[4] = {xpa, xpa, xpb, xpb};
  const int obase[4] = {EMB, 2 * EMB, EMB, 2 * EMB};
  for (int a = 0; a < 4; a++)
    edge_agg<<<gridW, blk, 0, stream>>>(dstv[a], ssrc + (size_t)a * N,
                                        sdst + (size_t)a * N, newemb[a],
                                        xdst[a], obase[a], N);

  // 6) fused update MLP via WMMA
  update_mlp<<<gridG, blk, 0, stream>>>(xpa, W1p, b1p, alpha, W2p, b2p,
                                        (float*)d_out, N);
  update_mlp<<<gridG, blk, 0, stream>>>(xpb, W1p, b1p, alpha, W2p, b2p,
                                        (float*)d_out + (size_t)N * EMB, N);
}